// KGCR_42408507081091
// MI455X (gfx1250) — compile-verified
//
#include <hip/hip_runtime.h>
#include <math.h>

// ---------------- problem constants (match reference) ----------------
#define NUM_U   100000
#define NUM_I   200000
#define DIM     64
#define N_UI    (NUM_U + NUM_I)   // 300000
#define N_IA    (2 * NUM_U)       // 200000
#define UI_E    2000000
#define IA_E    1000000
#define NPAIR   16384
#define ALPHA_C 0.2f
#define MARGIN_C 0.2f

typedef float v2f __attribute__((ext_vector_type(2)));
typedef float v8f __attribute__((ext_vector_type(8)));

// ---------------- utility kernels ----------------
__global__ void zero_f32(float* __restrict__ p, long long n) {
  long long i = (long long)blockIdx.x * blockDim.x + threadIdx.x;
  long long stride = (long long)gridDim.x * blockDim.x;
  for (; i < n; i += stride) p[i] = 0.0f;
}

__global__ void add_inplace_f4(float* __restrict__ a, const float* __restrict__ b, long long n4) {
  long long i = (long long)blockIdx.x * blockDim.x + threadIdx.x;
  long long stride = (long long)gridDim.x * blockDim.x;
  for (; i < n4; i += stride) {
    float4 x = ((float4*)a)[i];
    float4 y = ((const float4*)b)[i];
    x.x += y.x; x.y += y.y; x.z += y.z; x.w += y.w;
    ((float4*)a)[i] = x;
  }
}

// one wave (32 lanes) per edge; lane covers dims [2*lane, 2*lane+1]
// msg = w[e] * src[fold(row[e])] ; atomic-add into dst[col[e]]
// if cnt != nullptr, lane 0 adds 1.0 to cnt[col[e]]
__global__ void gcn_scatter(const float* __restrict__ src,
                            float* __restrict__ dst,
                            const int* __restrict__ row,
                            const int* __restrict__ col,
                            const float* __restrict__ w,
                            float* __restrict__ cnt,
                            int nedges, int srcFold) {
  int tid  = blockIdx.x * blockDim.x + threadIdx.x;
  int e    = tid >> 5;
  int lane = tid & 31;
  if (e >= nedges) return;
  int s = row[e];
  int d = col[e];
  float wt = w ? w[e] : 1.0f;
  if (srcFold && s >= srcFold) s -= srcFold;
  const float2 v = *(const float2*)(src + (long long)s * DIM + lane * 2);
  float* out = dst + (long long)d * DIM + lane * 2;
  atomicAdd(out + 0, v.x * wt);
  atomicAdd(out + 1, v.y * wt);
  if (cnt && lane == 0) atomicAdd(cnt + d, 1.0f);
}

// sum of squares -> atomicAdd into accum slot
__global__ void sumsq(const float* __restrict__ x, long long n, float* __restrict__ accum) {
  __shared__ float sdata[256];
  long long i = (long long)blockIdx.x * blockDim.x + threadIdx.x;
  long long stride = (long long)gridDim.x * blockDim.x;
  float s = 0.0f;
  for (; i < n; i += stride) { float v = x[i]; s += v * v; }
  sdata[threadIdx.x] = s;
  __syncthreads();
  for (int off = blockDim.x >> 1; off > 0; off >>= 1) {
    if ((int)threadIdx.x < off) sdata[threadIdx.x] += sdata[threadIdx.x + off];
    __syncthreads();
  }
  if (threadIdx.x == 0) atomicAdd(accum, sdata[0]);
}

// ---------------- WMMA pair-dot + loss kernel ----------------
#define WPB 4  // waves per block

// Fill 16 rows x 64 dims into LDS.
//  type 0: ui_rep  = 0.25*(id[idx] + Afold[idx])           (idx in [0,N_UI))
//  type 1: ia_rep  = 0.5 *(user_pre[idx] + Dbuf[idx])      (idx in [0,NUM_U))
//  type 2: hat_u   = Ebuf[idx] / max(Fcnt[idx],1)          (idx in [0,N_IA))
__device__ __forceinline__ void fill_rows(float* __restrict__ dst,
                                          const int* __restrict__ IDX, int idxBase, int sub,
                                          int type, int lane,
                                          const float* __restrict__ id_emb,
                                          const float* __restrict__ Afold,
                                          const float* __restrict__ user_pre,
                                          const float* __restrict__ Dbuf,
                                          const float* __restrict__ Ebuf,
                                          const float* __restrict__ Fcnt) {
  for (int t = lane; t < 16 * 64; t += 32) {
    int r = t >> 6, d = t & 63;
    long long idx = (long long)(IDX[idxBase + r] - sub);
    float v;
    if (type == 0)      v = 0.25f * (id_emb[idx * DIM + d] + Afold[idx * DIM + d]);
    else if (type == 1) v = 0.50f * (user_pre[idx * DIM + d] + Dbuf[idx * DIM + d]);
    else                v = Ebuf[idx * DIM + d] / fmaxf(Fcnt[idx], 1.0f);
    dst[t] = v;
  }
}

// 16 row-dots via V_WMMA_F32_16X16X4_F32 (full f32): accumulate U(16x64) * I(16x64)^T,
// then extract the diagonal per the 16x16 f32 C/D layout.
__device__ __forceinline__ void wmma_dots(const float* __restrict__ U,
                                          const float* __restrict__ I,
                                          float* __restrict__ dot16, int lane) {
  v8f c = {};
  const int mn = lane & 15;            // A: row M, B: col N (same lane mapping)
  const int kb = (lane >> 4) * 2;      // lanes 0-15: K=0,1 ; lanes 16-31: K=2,3
#pragma unroll
  for (int k = 0; k < 16; ++k) {
    v2f a, b;
    a.x = U[mn * 64 + 4 * k + kb + 0];
    a.y = U[mn * 64 + 4 * k + kb + 1];
    b.x = I[mn * 64 + 4 * k + kb + 0];   // B[k][n] = I[n][k] (I^T)
    b.y = I[mn * 64 + 4 * k + kb + 1];
    c = __builtin_amdgcn_wmma_f32_16x16x4_f32(false, a, false, b, (short)0, c, false, false);
  }
  // C layout: lanes 0-15 -> N=lane, VGPR j -> M=j ; lanes 16-31 -> N=lane-16, VGPR j -> M=j+8
  int want = (lane < 8) ? lane : ((lane >= 24) ? (lane - 24) : -1);
  float dv = 0.0f;
#pragma unroll
  for (int j = 0; j < 8; ++j) if (j == want) dv = c[j];
  if (lane < 8)        dot16[lane] = dv;
  else if (lane >= 24) dot16[lane - 16] = dv;
}

__global__ void dot_loss(const float* __restrict__ id_emb,
                         const float* __restrict__ user_pre,
                         const float* __restrict__ Afold,   // r1+r2+r3 (N_UI x 64)
                         const float* __restrict__ Dbuf,    // ia_rep_1 (N_IA x 64)
                         const float* __restrict__ Ebuf,    // agg      (N_IA x 64)
                         const float* __restrict__ Fcnt,    // counts   (N_IA)
                         const int* __restrict__ user_t,
                         const int* __restrict__ item_t,
                         float* __restrict__ accum) {       // [0]=sum l1, [1]=sum l2
  __shared__ float lds_u[WPB][16 * 64];
  __shared__ float lds_i[WPB][16 * 64];
  __shared__ int   lds_idx[WPB][32];        // [0..15]=user, [16..31]=item
  __shared__ float lds_dot[WPB][3][16];     // cf, kg, hat

  const int wave = threadIdx.x >> 5;
  const int lane = threadIdx.x & 31;
  const int g    = blockIdx.x * WPB + wave; // group of 16 pairs, 0..1023

  if (lane < 16) {
    lds_idx[wave][lane]      = user_t[g * 16 + lane];
    lds_idx[wave][lane + 16] = item_t[g * 16 + lane];
  }
  __syncthreads();

  float* U = lds_u[wave];
  float* I = lds_i[wave];
  const int* IDX = lds_idx[wave];

  // Round A: cf = <ui_rep[user], ui_rep[item]>
  fill_rows(U, IDX, 0, 0, 0, lane, id_emb, Afold, user_pre, Dbuf, Ebuf, Fcnt);
  fill_rows(I, IDX, 16, 0, 0, lane, id_emb, Afold, user_pre, Dbuf, Ebuf, Fcnt);
  __syncthreads();
  wmma_dots(U, I, lds_dot[wave][0], lane);
  __syncthreads();

  // Round B: kg = <ia_rep[user], ui_rep[item - NUM_U]>
  fill_rows(U, IDX, 0, 0, 1, lane, id_emb, Afold, user_pre, Dbuf, Ebuf, Fcnt);
  fill_rows(I, IDX, 16, NUM_U, 0, lane, id_emb, Afold, user_pre, Dbuf, Ebuf, Fcnt);
  __syncthreads();
  wmma_dots(U, I, lds_dot[wave][1], lane);
  __syncthreads();

  // Round C: hat = <all_hat_u[user], ui_rep[item - NUM_U]>  (I unchanged)
  fill_rows(U, IDX, 0, 0, 2, lane, id_emb, Afold, user_pre, Dbuf, Ebuf, Fcnt);
  __syncthreads();
  wmma_dots(U, I, lds_dot[wave][2], lane);
  __syncthreads();

  // loss partials: lanes 0..7 each handle one (pos,neg) pair
  float l1 = 0.0f, l2 = 0.0f;
  if (lane < 8) {
    const float(*DOT)[16] = lds_dot[wave];
    float cfp = DOT[0][2 * lane], cfn = DOT[0][2 * lane + 1];
    float kgp = DOT[1][2 * lane], kgn = DOT[1][2 * lane + 1];
    float htp = DOT[2][2 * lane], htn = DOT[2][2 * lane + 1];
    float sgp = 1.0f / (1.0f + expf(-htp));
    float sgn = 1.0f / (1.0f + expf(-htn));
    float pos = cfp + kgp * sgp;
    float neg = cfn + kgn * sgn;
    float score = 1.0f / (1.0f + expf(-(pos - neg)));
    l1 = -logf(fmaxf(score, 1e-6f));
    float sp = fmaxf(sgp, 1e-6f);
    float sn = fmaxf(sgn, 1e-6f);
    l2 = fmaxf(sp - sn - MARGIN_C, 0.0f);
  }
#pragma unroll
  for (int off = 16; off > 0; off >>= 1) {
    l1 += __shfl_down(l1, off, 32);
    l2 += __shfl_down(l2, off, 32);
  }
  if (lane == 0) {
    atomicAdd(accum + 0, l1);
    atomicAdd(accum + 1, l2);
  }
}

__global__ void finalize(const float* __restrict__ accum, float* __restrict__ out) {
  if (threadIdx.x == 0 && blockIdx.x == 0) {
    float loss1 = accum[0] / (float)(NPAIR / 2);
    float loss2 = accum[1] / (float)(NPAIR / 2);
    float reg = accum[2] / ((float)N_UI * DIM)
              + accum[3] / ((float)NUM_U * DIM)
              + accum[4] / ((float)NUM_I * DIM);
    out[0] = loss1 + ALPHA_C * loss2;
    out[1] = reg;
  }
}

// ---------------- driver ----------------
extern "C" void kernel_launch(void* const* d_in, const int* in_sizes, int n_in,
                              void* d_out, int out_size, void* d_ws, size_t ws_size,
                              hipStream_t stream) {
  const float* id_emb   = (const float*)d_in[0];
  const float* user_pre = (const float*)d_in[1];
  const float* item_pre = (const float*)d_in[2];
  const float* ui_w     = (const float*)d_in[3];
  const float* ia_w     = (const float*)d_in[4];
  // d_in[5] = ua_w : dead code in the reference (ua_rep unused)
  const int* user_t = (const int*)d_in[6];
  const int* item_t = (const int*)d_in[7];
  const int* ui_row = (const int*)d_in[8];
  const int* ui_col = (const int*)d_in[9];
  const int* ia_row = (const int*)d_in[10];
  const int* ia_col = (const int*)d_in[11];
  // d_in[12], d_in[13] = ua_row/ua_col : dead

  // workspace layout
  char* ws = (char*)d_ws;
  size_t off = 0;
  float* A = (float*)(ws + off); off += (size_t)N_UI * DIM * sizeof(float); // r1 -> r1+r2+r3
  float* B = (float*)(ws + off); off += (size_t)N_UI * DIM * sizeof(float); // r2
  float* D = (float*)(ws + off); off += (size_t)N_IA * DIM * sizeof(float); // ia_rep_1
  float* E = (float*)(ws + off); off += (size_t)N_IA * DIM * sizeof(float); // agg
  float* F = (float*)(ws + off); off += (size_t)N_IA * sizeof(float);       // counts
  float* accum = (float*)(ws + off); off += 64;                             // reductions
  (void)ws_size; (void)in_sizes; (void)n_in; (void)out_size; (void)item_pre;

  const long long nui = (long long)N_UI * DIM;
  const long long nia = (long long)N_IA * DIM;

  zero_f32<<<2048, 256, 0, stream>>>(A, nui);
  zero_f32<<<2048, 256, 0, stream>>>(B, nui);
  zero_f32<<<2048, 256, 0, stream>>>(D, nia);
  zero_f32<<<2048, 256, 0, stream>>>(E, nia);
  zero_f32<<<512, 256, 0, stream>>>(F, N_IA);
  zero_f32<<<1, 32, 0, stream>>>(accum, 8);

  const int tpb = 256;
  const int ui_blocks = (UI_E * 32) / tpb; // exact
  const int ia_blocks = (IA_E * 32) / tpb; // exact

  // r1 = gcn(id_embedding)
  gcn_scatter<<<ui_blocks, tpb, 0, stream>>>(id_emb, A, ui_row, ui_col, ui_w, nullptr, UI_E, 0);
  // r2 = gcn(r1)
  gcn_scatter<<<ui_blocks, tpb, 0, stream>>>(A, B, ui_row, ui_col, ui_w, nullptr, UI_E, 0);
  // A = r1 + r2
  add_inplace_f4<<<2048, 256, 0, stream>>>(A, B, nui / 4);
  // A += r3 = gcn(r2)   (scatter on top of non-zero base)
  gcn_scatter<<<ui_blocks, tpb, 0, stream>>>(B, A, ui_row, ui_col, ui_w, nullptr, UI_E, 0);

  // ia_rep_1 = gcn(concat(user_pre,user_pre)) ; source row folded mod NUM_U
  gcn_scatter<<<ia_blocks, tpb, 0, stream>>>(user_pre, D, ia_row, ia_col, ia_w, nullptr, IA_E, NUM_U);
  // agg = segment_sum(ia_rep_0[ia_col], ia_row) ; cnt = segment_sum(1, ia_row)
  gcn_scatter<<<ia_blocks, tpb, 0, stream>>>(user_pre, E, ia_col, ia_row, nullptr, F, IA_E, NUM_U);

  // reg_loss sums of squares
  sumsq<<<2048, 256, 0, stream>>>(id_emb, nui, accum + 2);
  sumsq<<<2048, 256, 0, stream>>>(user_pre, (long long)NUM_U * DIM, accum + 3);
  sumsq<<<2048, 256, 0, stream>>>(item_pre, (long long)NUM_I * DIM, accum + 4);

  // WMMA pair dots + loss partials: 1024 waves, 4 per block
  dot_loss<<<(NPAIR / 16) / WPB, 32 * WPB, 0, stream>>>(id_emb, user_pre, A, D, E, F,
                                                        user_t, item_t, accum);
  finalize<<<1, 1, 0, stream>>>(accum, (float*)d_out);
}